// GHM_Loss_Base_38878043963709
// MI455X (gfx1250) — compile-verified
//
#include <hip/hip_runtime.h>
#include <stdint.h>

// GHM-C loss for MI455X (gfx1250, wave32).
// Pass 1: masked histogram of bin(|sigmoid(p)-g|).
//         - p/g streamed via ASYNC global->LDS (global_load_async_to_lds_b128,
//           double-buffered, s_wait_asynccnt) -- CDNA5 async datapath, zero
//           VGPR pressure for in-flight loads, per-wave waits only (each lane
//           consumes exactly the bytes it loaded).
//         - 32-way lane-replicated LDS histogram: bank-conflict-free under
//           wave32, non-returning ds_add_u32 atomics, integer => deterministic.
// Pass 2: tiny 1-wave kernel computes per-bin weights.
// Pass 3: elementwise weighted BCE; p/g re-read (L2-resident: 128MB < 192MB L2)
//         with NT last-use loads; output streamed with NT stores.

#define MAX_BINS 64              // LDS sized for up to 64 bins; nb read on device
#define F16_EPS  0.0009765625f   // np.finfo(float16).eps
#define BLOCK    256             // 8 wave32s

typedef float float4v __attribute__((ext_vector_type(4)));
typedef __attribute__((address_space(3))) uint8_t lds_u8_t;

__device__ __forceinline__ float sigmoidf_fast(float x) {
    return 1.0f / (1.0f + __expf(-x));
}

// generic (flat) shared pointer -> 32-bit LDS byte offset (AS3 ptrs are 32-bit)
__device__ __forceinline__ uint32_t lds_off(const void* p) {
    return (uint32_t)(uintptr_t)(lds_u8_t*)(p);
}

// CDNA5 async copy: 16 bytes global -> LDS, tracked on ASYNCcnt (not LOADcnt)
__device__ __forceinline__ void async_load_b128(uint32_t lds_addr, const float4v* gptr) {
    asm volatile("global_load_async_to_lds_b128 %0, %1, off"
                 :: "v"(lds_addr), "v"(gptr)
                 : "memory");
}
__device__ __forceinline__ void wait_async_le(int) = delete;
__device__ __forceinline__ void wait_async_le2() {
    asm volatile("s_wait_asynccnt 2" ::: "memory");
}
__device__ __forceinline__ void wait_async_le0() {
    asm volatile("s_wait_asynccnt 0" ::: "memory");
}

// ---------------------------------------------------------------- init -------
__global__ void ghm_init_kernel(uint32_t* __restrict__ cnt) {
    if (threadIdx.x < MAX_BINS) cnt[threadIdx.x] = 0u;
}

// ------------------------------------------------------------- histogram -----
__global__ __launch_bounds__(BLOCK)
void ghm_hist_kernel(const float*   __restrict__ p,
                     const float*   __restrict__ g,
                     const uint8_t* __restrict__ mask,
                     const int*     __restrict__ nbins_p,
                     uint32_t*      __restrict__ gcnt,
                     long long n) {
    // lane-replicated histogram: lane L hits bank (32*bin+L)%64 -> no conflicts
    __shared__ uint32_t sh[MAX_BINS * 32];
    // double-buffered async staging: each thread owns 16B per buffer per array
    __shared__ float sp[2][BLOCK * 4];
    __shared__ float sg[2][BLOCK * 4];

    const int nb   = *nbins_p;
    const int lane = threadIdx.x & 31;
    const int t4   = threadIdx.x * 4;

    for (int i = threadIdx.x; i < MAX_BINS * 32; i += blockDim.x) sh[i] = 0u;
    __syncthreads();

    const float scale = (float)nb - F16_EPS;
    const long long n4     = n >> 2;
    const long long tid    = (long long)blockIdx.x * blockDim.x + threadIdx.x;
    const long long stride = (long long)gridDim.x * blockDim.x;

    const float4v*  p4 = (const float4v*)p;
    const float4v*  g4 = (const float4v*)g;
    const uint32_t* m4 = (const uint32_t*)mask;

    // ---- async double-buffered pipeline over float4 tiles ----
    int buf = 0;
    if (tid < n4) {
        async_load_b128(lds_off(&sp[0][t4]), &p4[tid]);
        async_load_b128(lds_off(&sg[0][t4]), &g4[tid]);
    }
    for (long long i = tid; i < n4; i += stride) {
        const long long inext = i + stride;
        if (inext < n4) {
            async_load_b128(lds_off(&sp[buf ^ 1][t4]), &p4[inext]);
            async_load_b128(lds_off(&sg[buf ^ 1][t4]), &g4[inext]);
            wait_async_le2();            // current tile's 2 loads complete,
                                         // next tile's 2 stay in flight
        } else {
            wait_async_le0();
        }
        uint32_t mv = __builtin_nontemporal_load(&m4[i]);  // mask read once -> NT
        float4v  pv = *(const float4v*)&sp[buf][t4];        // ds_load_b128
        float4v  gv = *(const float4v*)&sg[buf][t4];

#pragma unroll
        for (int c = 0; c < 4; ++c) {
            if ((mv >> (8 * c)) & 0xFFu) {
                float glen = fabsf(sigmoidf_fast(pv[c]) - gv[c]);
                int   b    = (int)(glen * scale);
                b = b < 0 ? 0 : (b >= nb ? nb - 1 : b);
                atomicAdd(&sh[b * 32 + lane], 1u);          // ds_add_u32
            }
        }
        buf ^= 1;
    }
    // scalar tail (n not multiple of 4)
    for (long long i = (n4 << 2) + tid; i < n; i += stride) {
        if (mask[i]) {
            float glen = fabsf(sigmoidf_fast(p[i]) - g[i]);
            int   b    = (int)(glen * scale);
            b = b < 0 ? 0 : (b >= nb ? nb - 1 : b);
            atomicAdd(&sh[b * 32 + lane], 1u);
        }
    }
    __syncthreads();

    // fold 32 lane replicas, one global atomic per (block, bin)
    for (int b = threadIdx.x; b < nb; b += blockDim.x) {
        uint32_t s = 0;
#pragma unroll
        for (int l = 0; l < 32; ++l) s += sh[b * 32 + l];
        if (s) atomicAdd(&gcnt[b], s);                      // integer -> deterministic
    }
}

// ------------------------------------------------------------- finalize ------
__global__ void ghm_finalize_kernel(const uint32_t* __restrict__ cnt,
                                    const int*      __restrict__ nbins_p,
                                    float*          __restrict__ wts) {
    if (threadIdx.x == 0) {
        const int nb = *nbins_p;
        float num_calc = 0.0f, valid = 0.0f;
        for (int b = 0; b < nb; ++b) {
            float c = (float)cnt[b];
            num_calc += c;                 // sum(mask) == histogram total
            if (c > 0.0f) valid += 1.0f;
        }
        for (int b = 0; b < nb; ++b) {
            float c = (float)cnt[b] / valid;
            wts[b]  = num_calc / fmaxf(c, F16_EPS);
        }
    }
}

// ---------------------------------------------------------------- apply ------
__global__ __launch_bounds__(BLOCK)
void ghm_apply_kernel(const float* __restrict__ p,
                      const float* __restrict__ g,
                      const int*   __restrict__ nbins_p,
                      const float* __restrict__ wts,
                      float*       __restrict__ out,
                      long long n) {
    __shared__ float swt[MAX_BINS];
    const int nb = *nbins_p;
    for (int i = threadIdx.x; i < nb; i += blockDim.x) swt[i] = wts[i];
    __syncthreads();

    const float scale = (float)nb - F16_EPS;
    const long long n4     = n >> 2;
    const long long tid    = (long long)blockIdx.x * blockDim.x + threadIdx.x;
    const long long stride = (long long)gridDim.x * blockDim.x;

    const float4v* p4 = (const float4v*)p;
    const float4v* g4 = (const float4v*)g;
    float4v*       o4 = (float4v*)out;

    for (long long i = tid; i < n4; i += stride) {
        // last use of p/g: NT loads (served from the L2-resident copy)
        float4v pv = __builtin_nontemporal_load(&p4[i]);
        float4v gv = __builtin_nontemporal_load(&g4[i]);
        float4v ov;
#pragma unroll
        for (int c = 0; c < 4; ++c) {
            float pc = pv[c], gc = gv[c];
            float glen = fabsf(sigmoidf_fast(pc) - gc);
            int   b    = (int)(glen * scale);
            b = b < 0 ? 0 : (b >= nb ? nb - 1 : b);
            float bce = fmaxf(pc, 0.0f) - pc * gc + log1pf(__expf(-fabsf(pc)));
            ov[c] = swt[b] * bce;                           // ds_load weight gather
        }
        __builtin_nontemporal_store(ov, &o4[i]);            // streaming NT store
    }
    for (long long i = (n4 << 2) + tid; i < n; i += stride) {
        float pc = p[i], gc = g[i];
        float glen = fabsf(sigmoidf_fast(pc) - gc);
        int   b    = (int)(glen * scale);
        b = b < 0 ? 0 : (b >= nb ? nb - 1 : b);
        float bce = fmaxf(pc, 0.0f) - pc * gc + log1pf(__expf(-fabsf(pc)));
        out[i] = swt[b] * bce;
    }
}

// ---------------------------------------------------------------- launch -----
extern "C" void kernel_launch(void* const* d_in, const int* in_sizes, int n_in,
                              void* d_out, int out_size, void* d_ws, size_t ws_size,
                              hipStream_t stream) {
    const float*   pconf = (const float*)d_in[0];
    const float*   gconf = (const float*)d_in[1];
    const uint8_t* mask  = (const uint8_t*)d_in[2];   // jnp.bool_: 1 byte/elem
    const int*     nbp   = (const int*)d_in[3];       // num_bins (device scalar)
    float*         out   = (float*)d_out;

    const long long n = (long long)in_sizes[0];

    // workspace layout: [0..63] u32 bin counts, [64..127] f32 bin weights
    uint32_t* cnt = (uint32_t*)d_ws;
    float*    wts = (float*)d_ws + MAX_BINS;

    long long n4 = (n + 3) >> 2;
    int grid = (int)((n4 + BLOCK - 1) / BLOCK);
    if (grid > 4096) grid = 4096;   // grid-stride: exactly 4 uniform iters @16M
    if (grid < 1)    grid = 1;

    ghm_init_kernel<<<1, 64, 0, stream>>>(cnt);
    ghm_hist_kernel<<<grid, BLOCK, 0, stream>>>(pconf, gconf, mask, nbp, cnt, n);
    ghm_finalize_kernel<<<1, 32, 0, stream>>>(cnt, nbp, wts);
    ghm_apply_kernel<<<grid, BLOCK, 0, stream>>>(pconf, gconf, nbp, wts, out, n);
}